// FastAttention_58488864637348
// MI455X (gfx1250) — compile-verified
//
#include <hip/hip_runtime.h>
#include <math.h>

// Problem constants (B=4,H=8,N=4096,D=64,M=266, chunk=128)
#define BDIM 4
#define HDIM 8
#define NDIM 4096
#define DDIM 64
#define MREAL 266
#define MP 272            // padded M (17 * 16)
#define MT (MP/16)        // 17 column tiles
#define CHK 128
#define NCH (NDIM/CHK)    // 32 chunks
#define BH (BDIM*HDIM)    // 32
#define ROWS ((size_t)BH*NDIM)  // 131072

#define NORMALIZER 0.35355339059327373f   // 64^-0.25
#define RATIO      0.061313923f           // 266^-0.5
#define KEPS       1e-4f

typedef __attribute__((ext_vector_type(2))) float v2f;
typedef __attribute__((ext_vector_type(8))) float v8f;
typedef __attribute__((ext_vector_type(4))) unsigned int v4u;
typedef __attribute__((ext_vector_type(8))) int v8i;
typedef __attribute__((ext_vector_type(4))) int v4i;

__device__ __forceinline__ v8f wmma4(v2f a, v2f b, v8f c) {
  // D = A(16x4 f32) * B(4x16 f32) + C(16x16 f32), full f32 precision
  return __builtin_amdgcn_wmma_f32_16x16x4_f32(
      false, a, false, b, (short)0, c, false, false);
}

// Tensor Data Mover: DMA a [height x width] f32 tile (row stride in elems)
// from global memory into contiguous LDS at lds_off.  D# packing per CDNA5
// ISA ch.8 (group0: count/lds_addr/global_addr/type=2; group1: data_size=4B,
// tensor dims, tile dims, dim0 stride; remaining groups zero => 2-D tensor).
// This toolchain exposes the 6-arg builtin: (v4u, v8i, v4i, v4i, v8i, i32 cpol).
__device__ __forceinline__ void tdm_load_2d(const void* gptr, unsigned lds_off,
                                            unsigned width, unsigned height,
                                            unsigned stride) {
  unsigned long long ga = (unsigned long long)gptr;
  v4u g0;
  g0.x = 1u;                                            // count=1 (user D#)
  g0.y = lds_off;                                       // lds_addr (bytes)
  g0.z = (unsigned)(ga & 0xffffffffu);                  // global_addr[31:0]
  g0.w = (unsigned)((ga >> 32) & 0x01ffffffu) | (2u << 30); // addr[56:32]|type=2
  v8i g1;
  g1[0] = (int)(2u << 16);                              // data_size=2 (4 bytes)
  g1[1] = (int)((width & 0xffffu) << 16);               // tensor_dim0[15:0]
  g1[2] = (int)((width >> 16) | ((height & 0xffffu) << 16)); // dim0 hi | tensor_dim1 lo
  g1[3] = (int)((height >> 16) | (width << 16));        // tensor_dim1 hi | tile_dim0
  g1[4] = (int)(height & 0xffffu);                      // tile_dim1 (tile_dim2=0)
  g1[5] = (int)stride;                                  // tensor_dim0_stride[31:0]
  g1[6] = 0;                                            // stride hi | dim1_stride lo
  g1[7] = 0;
  v4i z4 = {0, 0, 0, 0};
  v8i z8 = {0, 0, 0, 0, 0, 0, 0, 0};
  __builtin_amdgcn_tensor_load_to_lds(g0, g1, z4, z4, z8, 0);
}

__device__ __forceinline__ unsigned lds_off(const void* p) {
  return (unsigned)(unsigned long long)(uintptr_t)p;    // low 32 bits = LDS offset
}

// ---------------------------------------------------------------------------
// Kernel 1: projection  dd = (normalizer*X) @ P^T   (X:[rows,64], P:[266,64])
// IS_Q: finalize qp = ratio*(exp(dd - diag - rowmax)+eps) directly.
// !IS_Q: store raw dd (padded cols zero), diag per row, per-block max.
// ---------------------------------------------------------------------------
template<bool IS_Q>
__global__ __launch_bounds__(128) void proj_kernel(
    const float* __restrict__ X, const float* __restrict__ P,
    float* __restrict__ outp, float* __restrict__ diagk,
    float* __restrict__ bmax) {
  __shared__ float xs[64*64];     // 16KB scaled X rows
  __shared__ float dd[64*MP];     // 69.6KB staging of projections
  __shared__ float ptile[16*64];  // 4KB streamed P tile
  __shared__ float red[64];
  const int tid = threadIdx.x;
  const size_t r0 = (size_t)blockIdx.x * 64;

  for (int i = tid; i < 64*64; i += 128)
    xs[i] = NORMALIZER * X[r0*64 + i];

  const int wave = tid >> 5, lane = tid & 31, half = lane >> 4, lm = lane & 15;
  const int arow = wave*16 + lm;  // wave handles rows [16w,16w+15], all col tiles

  for (int ct = 0; ct < MT; ++ct) {
    __syncthreads();
    for (int i = tid; i < 16*64; i += 128) {
      int m = ct*16 + (i >> 6);
      ptile[i] = (m < MREAL) ? P[(size_t)m*64 + (i & 63)] : 0.0f;
    }
    __syncthreads();
    v8f acc = {};
#pragma unroll
    for (int k0 = 0; k0 < 64; k0 += 4) {
      const int kk = k0 + 2*half;
      v2f a; a.x = xs[arow*64 + kk];  a.y = xs[arow*64 + kk + 1];
      v2f b; b.x = ptile[lm*64 + kk]; b.y = ptile[lm*64 + kk + 1];
      acc = wmma4(a, b, acc);
    }
#pragma unroll
    for (int i = 0; i < 8; ++i)
      dd[(wave*16 + i + 8*half)*MP + ct*16 + lm] = acc[i];
  }
  __syncthreads();

  if (tid < 64) {
    const size_t g = r0 + tid;
    float diag = 0.f;
    for (int kd = 0; kd < 64; ++kd) { float x = xs[tid*64 + kd]; diag += x*x; }
    diag *= 0.5f;
    float mx = -3.0e38f;
    for (int m = 0; m < MREAL; ++m) mx = fmaxf(mx, dd[tid*MP + m]);
    if (IS_Q) {
      for (int m = 0; m < MP; ++m) {
        float v = (m < MREAL)
            ? RATIO * (expf(dd[tid*MP + m] - diag - mx) + KEPS) : 0.f;
        outp[g*MP + m] = v;
      }
    } else {
      diagk[g] = diag;
      for (int m = 0; m < MP; ++m)
        outp[g*MP + m] = (m < MREAL) ? dd[tid*MP + m] : 0.f;
      red[tid] = mx;
    }
  }
  if (!IS_Q) {
    __syncthreads();
    if (tid == 0) {
      float m = red[0];
      for (int i = 1; i < 64; ++i) m = fmaxf(m, red[i]);
      bmax[blockIdx.x] = m;
    }
  }
}

// ---------------------------------------------------------------------------
// Kernel 2: reduce per-block maxima -> global scalar (deterministic)
// ---------------------------------------------------------------------------
__global__ __launch_bounds__(256) void reduce_max_kernel(
    const float* __restrict__ bmax, float* __restrict__ gmax, int n) {
  __shared__ float sb[256];
  float m = -3.0e38f;
  for (int i = threadIdx.x; i < n; i += 256) m = fmaxf(m, bmax[i]);
  sb[threadIdx.x] = m;
  __syncthreads();
  for (int s = 128; s > 0; s >>= 1) {
    if (threadIdx.x < s) sb[threadIdx.x] = fmaxf(sb[threadIdx.x], sb[threadIdx.x + s]);
    __syncthreads();
  }
  if (threadIdx.x == 0) *gmax = sb[0];
}

// ---------------------------------------------------------------------------
// Kernel 3: per (bh,chunk): finalize kp in place, S_c = K^T V (WMMA),
// ksum_c = column sums of kp chunk.  V chunk staged via TDM.
// ---------------------------------------------------------------------------
__global__ __launch_bounds__(128) void chunksum_kernel(
    float* __restrict__ kp, const float* __restrict__ V,
    const float* __restrict__ diagk, const float* __restrict__ gmaxp,
    float* __restrict__ S, float* __restrict__ ksum) {
  __shared__ float vs[CHK*64];   // 32KB
  __shared__ float kt[CHK*16];   // 8KB streamed kp tile
  __shared__ float kspart[8*16];
  const int tid = threadIdx.x;
  const int bh = blockIdx.x / NCH, ch = blockIdx.x % NCH;
  const size_t g0 = (size_t)bh*NDIM + (size_t)ch*CHK;
  const float gmax = *gmaxp;

  if (tid < 32) {  // wave 0: DMA the V chunk (contiguous 128x64 f32) into LDS
    tdm_load_2d(V + g0*64, lds_off(vs), CHK*64, 1, CHK*64);
    __builtin_amdgcn_s_wait_tensorcnt(0);
  }
  __syncthreads();

  const int wave = tid >> 5, lane = tid & 31, half = lane >> 4, lm = lane & 15;
  // preload B fragments (V columns owned by this wave); reused for every m tile
  v2f bf[32];
#pragma unroll
  for (int ks = 0; ks < 32; ++ks) {
    int kk = ks*4 + 2*half;
    bf[ks].x = vs[kk*64 + wave*16 + lm];
    bf[ks].y = vs[(kk+1)*64 + wave*16 + lm];
  }
  const size_t sbase = (size_t)blockIdx.x * MP * 64;

  for (int mt = 0; mt < MT; ++mt) {
    __syncthreads();
    for (int i = tid; i < CHK*16; i += 128) {
      int r = i >> 4, m = mt*16 + (i & 15);
      size_t gi = (g0 + r)*MP + m;
      float raw = kp[gi];
      float val = (m < MREAL)
          ? RATIO * (expf(raw - diagk[g0 + r] - gmax) + KEPS) : 0.f;
      kt[i] = val;
      kp[gi] = val;   // finalized kp, consumed by attention pass
    }
    __syncthreads();
    v8f acc = {};
#pragma unroll
    for (int ks = 0; ks < 32; ++ks) {
      int kk = ks*4 + 2*half;
      v2f a; a.x = kt[kk*16 + lm]; a.y = kt[(kk+1)*16 + lm];
      acc = wmma4(a, bf[ks], acc);
    }
#pragma unroll
    for (int i = 0; i < 8; ++i) {
      int m = mt*16 + i + 8*half;
      S[sbase + (size_t)m*64 + wave*16 + lm] = acc[i];
    }
    // column sums of kp tile
    {
      int col = tid & 15, grp = tid >> 4;
      float p = 0.f;
      for (int r = 0; r < 16; ++r) p += kt[(grp*16 + r)*16 + col];
      kspart[grp*16 + col] = p;
    }
    __syncthreads();
    if (tid < 16) {
      float s = 0.f;
      for (int gi = 0; gi < 8; ++gi) s += kspart[gi*16 + tid];
      ksum[(size_t)blockIdx.x*MP + mt*16 + tid] = s;
    }
  }
}

// ---------------------------------------------------------------------------
// Kernel 4: exclusive prefix scan over the 32 chunks (per bh), in place.
// ---------------------------------------------------------------------------
__global__ __launch_bounds__(256) void scan_kernel(
    float* __restrict__ S, float* __restrict__ ksum) {
  const int bh = blockIdx.x, tid = threadIdx.x;
  float run[68];   // MP*64/256 = 68 elements per thread
#pragma unroll
  for (int j = 0; j < 68; ++j) run[j] = 0.f;
  for (int c = 0; c < NCH; ++c) {
    size_t base = ((size_t)bh*NCH + c) * ((size_t)MP*64);
#pragma unroll
    for (int j = 0; j < 68; ++j) {
      size_t idx = base + tid + j*256;
      float t = S[idx];
      S[idx] = run[j];
      run[j] += t;
    }
  }
  float r2[2] = {0.f, 0.f};
  for (int c = 0; c < NCH; ++c) {
    size_t base = ((size_t)bh*NCH + c) * MP;
#pragma unroll
    for (int j = 0; j < 2; ++j) {
      int m = tid + j*256;
      if (m < MP) {
        float t = ksum[base + m];
        ksum[base + m] = r2[j];
        r2[j] += t;
      }
    }
  }
}

// ---------------------------------------------------------------------------
// Kernel 5: per (bh,chunk): out = (tril(Q K^T) V + Q CTX) * Dinv
//           Dinv = 1/(Q.ksum_prev + rowsum(tril(Q K^T)))
// All tile staging via Tensor Data Mover (TENSORcnt + barrier).
// ---------------------------------------------------------------------------
__global__ __launch_bounds__(128) void attn_kernel(
    const float* __restrict__ qp, const float* __restrict__ kp,
    const float* __restrict__ V, const float* __restrict__ CTX,
    const float* __restrict__ ksum, float* __restrict__ out) {
  __shared__ float A_lds[CHK*CHK];  // 64KB masked QK^T
  __shared__ float vs[CHK*64];      // 32KB
  __shared__ float qt[CHK*16];      // 8KB
  __shared__ float kt[CHK*16];      // 8KB
  __shared__ float ctxt[16*64];     // 4KB
  __shared__ float ksumt[16];
  __shared__ float dinv[CHK];
  const int tid = threadIdx.x;
  const int bh = blockIdx.x / NCH, ch = blockIdx.x % NCH;
  const size_t g0 = (size_t)bh*NDIM + (size_t)ch*CHK;
  const int wave = tid >> 5, lane = tid & 31, half = lane >> 4, lm = lane & 15;

  v8f accA[2][8] = {};  // QK^T band: rows [32w,32w+31] x 128 cols
  v8f accO[2][4] = {};  // out band:  rows [32w,32w+31] x 64 cols
  float dprev = 0.f;
  const size_t cbase = (size_t)blockIdx.x * MP * 64;

  for (int mt = 0; mt < MT; ++mt) {
    __syncthreads();   // everyone done reading previous tiles
    if (tid < 32) {    // wave 0 drives the TDM for this m tile
      tdm_load_2d(qp + g0*MP + mt*16, lds_off(qt), 16, CHK, MP);   // [128x16], stride MP
      tdm_load_2d(kp + g0*MP + mt*16, lds_off(kt), 16, CHK, MP);
      tdm_load_2d(CTX + cbase + (size_t)mt*16*64, lds_off(ctxt), 16*64, 1, 16*64);
      tdm_load_2d(ksum + (size_t)blockIdx.x*MP + mt*16, lds_off(ksumt), 16, 1, 16);
      __builtin_amdgcn_s_wait_tensorcnt(0);
    }
    __syncthreads();   // publish DMA'd tiles to all waves
    // D contribution from previous chunks (thread == row)
    {
      float s = 0.f;
#pragma unroll
      for (int c = 0; c < 16; ++c) s += qt[tid*16 + c] * ksumt[c];
      dprev += s;
    }
#pragma unroll
    for (int tr = 0; tr < 2; ++tr) {
      const int row = (2*wave + tr)*16 + lm;
      v2f af[4];
#pragma unroll
      for (int ks = 0; ks < 4; ++ks) {
        int kk = ks*4 + 2*half;
        af[ks].x = qt[row*16 + kk];
        af[ks].y = qt[row*16 + kk + 1];
      }
#pragma unroll
      for (int tc = 0; tc < 8; ++tc) {
#pragma unroll
        for (int ks = 0; ks < 4; ++ks) {
          int kk = ks*4 + 2*half;
          v2f b; b.x = kt[(tc*16 + lm)*16 + kk]; b.y = kt[(tc*16 + lm)*16 + kk + 1];
          accA[tr][tc] = wmma4(af[ks], b, accA[tr][tc]);
        }
      }
#pragma unroll
      for (int tc = 0; tc < 4; ++tc) {
#pragma unroll
        for (int ks = 0; ks < 4; ++ks) {
          int kk = ks*4 + 2*half;
          v2f b; b.x = ctxt[kk*64 + tc*16 + lm]; b.y = ctxt[(kk+1)*64 + tc*16 + lm];
          accO[tr][tc] = wmma4(af[ks], b, accO[tr][tc]);
        }
      }
    }
  }
  __syncthreads();

  // DMA the V chunk while all waves spill the masked A matrix to LDS
  if (tid < 32)
    tdm_load_2d(V + g0*64, lds_off(vs), CHK*64, 1, CHK*64);
#pragma unroll
  for (int tr = 0; tr < 2; ++tr) {
#pragma unroll
    for (int tc = 0; tc < 8; ++tc) {
#pragma unroll
      for (int i = 0; i < 8; ++i) {
        int r = (2*wave + tr)*16 + i + 8*half;
        int c = tc*16 + lm;
        A_lds[r*CHK + c] = (r >= c) ? accA[tr][tc][i] : 0.f;
      }
    }
  }
  if (tid < 32) __builtin_amdgcn_s_wait_tensorcnt(0);
  __syncthreads();
  {
    float s = 0.f;
    for (int j = 0; j < CHK; ++j) s += A_lds[tid*CHK + j];
    dinv[tid] = 1.0f / (dprev + s);
  }
  __syncthreads();
  // out += tril(A) @ V
#pragma unroll
  for (int tr = 0; tr < 2; ++tr) {
    const int row = (2*wave + tr)*16 + lm;
    for (int ks = 0; ks < 32; ++ks) {
      int kk = ks*4 + 2*half;
      v2f a; a.x = A_lds[row*CHK + kk]; a.y = A_lds[row*CHK + kk + 1];
#pragma unroll
      for (int tc = 0; tc < 4; ++tc) {
        v2f b; b.x = vs[kk*64 + tc*16 + lm]; b.y = vs[(kk+1)*64 + tc*16 + lm];
        accO[tr][tc] = wmma4(a, b, accO[tr][tc]);
      }
    }
  }
#pragma unroll
  for (int tr = 0; tr < 2; ++tr) {
#pragma unroll
    for (int tc = 0; tc < 4; ++tc) {
#pragma unroll
      for (int i = 0; i < 8; ++i) {
        int r = (2*wave + tr)*16 + i + 8*half;
        int c = tc*16 + lm;
        out[(g0 + r)*64 + c] = accO[tr][tc][i] * dinv[r];
      }
    }
  }
}

// ---------------------------------------------------------------------------
extern "C" void kernel_launch(void* const* d_in, const int* in_sizes, int n_in,
                              void* d_out, int out_size, void* d_ws, size_t ws_size,
                              hipStream_t stream) {
  (void)in_sizes; (void)n_in; (void)out_size; (void)ws_size;
  const float* q = (const float*)d_in[0];
  const float* k = (const float*)d_in[1];
  const float* v = (const float*)d_in[2];
  const float* P = (const float*)d_in[3];
  float* out = (float*)d_out;

  float* ws   = (float*)d_ws;
  float* qp   = ws;                                   // ROWS*MP
  float* kp   = qp + ROWS*(size_t)MP;                 // ROWS*MP
  float* S    = kp + ROWS*(size_t)MP;                 // BH*NCH*MP*64
  float* ksum = S + (size_t)BH*NCH*MP*64;             // BH*NCH*MP
  float* diagk= ksum + (size_t)BH*NCH*MP;             // ROWS
  float* bmax = diagk + ROWS;                         // 2048
  float* gmax = bmax + 2048;                          // 1

  const int nblk = (int)(ROWS / 64);                  // 2048
  proj_kernel<true ><<<nblk, 128, 0, stream>>>(q, P, qp, nullptr, nullptr);
  proj_kernel<false><<<nblk, 128, 0, stream>>>(k, P, kp, diagk, bmax);
  reduce_max_kernel<<<1, 256, 0, stream>>>(bmax, gmax, nblk);
  chunksum_kernel<<<BH*NCH, 128, 0, stream>>>(kp, v, diagk, gmax, S, ksum);
  scan_kernel<<<BH, 256, 0, stream>>>(S, ksum);
  attn_kernel<<<BH*NCH, 128, 0, stream>>>(qp, kp, v, S, ksum, out);
}